// GraphPolicyNetwork_4612794876151
// MI455X (gfx1250) — compile-verified
//
#include <hip/hip_runtime.h>
#include <hip/hip_bf16.h>
#include <math.h>

typedef float v2f __attribute__((ext_vector_type(2)));
typedef float v8f __attribute__((ext_vector_type(8)));

#define NDIM 4096      // n_nodes per graph = GEMM N = GEMM K
#define KSPLIT 8       // K-chunks per N-strip for occupancy (512 blocks total)

__device__ __forceinline__ void atomAddF(float* p, float v) {
    __hip_atomic_fetch_add(p, v, __ATOMIC_RELAXED, __HIP_MEMORY_SCOPE_AGENT);
}

// ---------------- Pass 0: zero accumulators, seed out2 with bias -------------
__global__ void gpn_init_kernel(float* __restrict__ agg1,
                                float* __restrict__ deg,
                                float* __restrict__ agg2,
                                float* __restrict__ out2,
                                const float* __restrict__ b3, int N) {
    int n = blockIdx.x * blockDim.x + threadIdx.x;
    if (n < N) {
        agg1[n] = 0.0f; deg[n] = 0.0f; agg2[n] = 0.0f;
        out2[n] = b3[n & (NDIM - 1)];      // out2[row][col] = b3[col]
    }
}

// ---------------- Pass 1: edge scatter (layer-1 aggregation, scalar feats) ---
__global__ void gpn_edge1_kernel(const int* __restrict__ src,
                                 const int* __restrict__ dst,
                                 const float* __restrict__ feat,
                                 float* __restrict__ agg1,
                                 float* __restrict__ deg, int E) {
    int i = blockIdx.x * blockDim.x + threadIdx.x;
    int base = i * 4;
    if (base + 3 < E) {
        int4 s4 = ((const int4*)src)[i];
        int4 d4 = ((const int4*)dst)[i];
        atomAddF(&agg1[d4.x], feat[s4.x]);  atomAddF(&deg[d4.x], 1.0f);
        atomAddF(&agg1[d4.y], feat[s4.y]);  atomAddF(&deg[d4.y], 1.0f);
        atomAddF(&agg1[d4.z], feat[s4.z]);  atomAddF(&deg[d4.z], 1.0f);
        atomAddF(&agg1[d4.w], feat[s4.w]);  atomAddF(&deg[d4.w], 1.0f);
    } else {
        for (int e = base; e < E; ++e) {
            atomAddF(&agg1[dst[e]], feat[src[e]]);
            atomAddF(&deg[dst[e]], 1.0f);
        }
    }
}

// ---------------- Pass 2: per-node hidden layer, contracted with w2 ----------
// h1[n,j] = tanh(a*ws1[j] + c*wn1[j] + b1[j]); s_self = h1.ws2 (+b2), s_neigh = h1.wn2
__global__ void gpn_node1_kernel(const float* __restrict__ feat,
                                 const float* __restrict__ agg1,
                                 const float* __restrict__ deg,
                                 const float* __restrict__ ws1,
                                 const float* __restrict__ wn1,
                                 const float* __restrict__ b1,
                                 const float* __restrict__ ws2,
                                 const float* __restrict__ wn2,
                                 const float* __restrict__ b2,
                                 float* __restrict__ sself,
                                 float* __restrict__ sneigh,
                                 int N, int H) {
    int n = blockIdx.x * blockDim.x + threadIdx.x;
    if (n >= N) return;
    float a  = feat[n];
    float dg = deg[n];
    float c  = agg1[n] / fmaxf(dg, 1.0f);
    float ss = 0.0f, sn = 0.0f;
#pragma unroll 4
    for (int j = 0; j < H; ++j) {
        float z = tanhf(fmaf(a, ws1[j], fmaf(c, wn1[j], b1[j])));
        ss = fmaf(z, ws2[j], ss);
        sn = fmaf(z, wn2[j], sn);
    }
    sself[n]  = ss + b2[0];   // fold b2 here
    sneigh[n] = sn;
}

// ---------------- Pass 3: edge scatter (layer-2, reduced to scalars) ---------
__global__ void gpn_edge2_kernel(const int* __restrict__ src,
                                 const int* __restrict__ dst,
                                 const float* __restrict__ sneigh,
                                 float* __restrict__ agg2, int E) {
    int i = blockIdx.x * blockDim.x + threadIdx.x;
    int base = i * 4;
    if (base + 3 < E) {
        int4 s4 = ((const int4*)src)[i];
        int4 d4 = ((const int4*)dst)[i];
        atomAddF(&agg2[d4.x], sneigh[s4.x]);
        atomAddF(&agg2[d4.y], sneigh[s4.y]);
        atomAddF(&agg2[d4.z], sneigh[s4.z]);
        atomAddF(&agg2[d4.w], sneigh[s4.w]);
    } else {
        for (int e = base; e < E; ++e) atomAddF(&agg2[dst[e]], sneigh[src[e]]);
    }
}

// ---------------- Pass 4: out1 = s_self + mean(agg2); h2 = tanh(out1) --------
__global__ void gpn_node2_kernel(const float* __restrict__ sself,
                                 const float* __restrict__ agg2,
                                 const float* __restrict__ deg,
                                 float* __restrict__ out1,
                                 float* __restrict__ h2, int N) {
    int n = blockIdx.x * blockDim.x + threadIdx.x;
    if (n >= N) return;
    float o = sself[n] + agg2[n] / fmaxf(deg[n], 1.0f);
    out1[n] = o;
    h2[n]   = tanhf(o);
}

// ---------------- Pass 5: output2 += h2[64,4096] @ w3[4096,4096] -------------
// fp32 WMMA 16x16x4. Each wave owns one 16-wide N strip x one K-chunk; all 4
// M-tiles (M=64) live in 32 accumulator VGPRs so w3 streams from HBM exactly
// once chip-wide. A (h2) staged in LDS in 64-wide K chunks. K is split
// KSPLIT ways across blockIdx.y for occupancy; partials accumulate into out2
// (pre-seeded with b3) via GLOBAL_ATOMIC_ADD_F32.
__global__ __launch_bounds__(128) void gpn_gemm_kernel(
        const float* __restrict__ h2,
        const float* __restrict__ w3,
        float* __restrict__ out2) {
    __shared__ float Alds[64 * 64];               // 16 KB: 64 rows x 64 k
    const int t    = threadIdx.x;
    const int wave = t >> 5;
    const int lane = t & 31;
    const int ln   = lane & 15;                   // N / M position within tile
    const int kb   = (lane >> 4) << 1;            // K sub-offset: 0 or 2
    const int n0   = (blockIdx.x * 4 + wave) * 16;
    const int kc0  = blockIdx.y * (NDIM / KSPLIT);
    const int kc1  = kc0 + (NDIM / KSPLIT);

    v8f acc[4];
#pragma unroll
    for (int m = 0; m < 4; ++m)
#pragma unroll
        for (int v = 0; v < 8; ++v) acc[m][v] = 0.0f;

    for (int kc = kc0; kc < kc1; kc += 64) {
        __syncthreads();                          // protect prior chunk reads
        // cooperative, coalesced A-chunk load: h2[row][kc+kk]
        for (int e = t; e < 64 * 64; e += 128) {
            int row = e >> 6, kk = e & 63;
            Alds[e] = h2[row * NDIM + kc + kk];
        }
        __syncthreads();
#pragma unroll 4
        for (int kk = 0; kk < 64; kk += 4) {
            // B frag: 4x16 strip of w3 (rows kc+kk..+3, cols n0..n0+15)
            size_t krow = (size_t)(kc + kk + kb);
            v2f bfrag;
            bfrag.x = w3[krow * NDIM + n0 + ln];
            bfrag.y = w3[(krow + 1) * NDIM + n0 + ln];
#pragma unroll
            for (int m = 0; m < 4; ++m) {
                int r = m * 16 + ln;              // A row for this lane
                v2f afrag = *(const v2f*)&Alds[r * 64 + kk + kb];
                acc[m] = __builtin_amdgcn_wmma_f32_16x16x4_f32(
                    false, afrag, false, bfrag, (short)0, acc[m], false, false);
            }
        }
    }
    // D layout: VGPR v, lanes 0-15 -> M=v, N=lane; lanes 16-31 -> M=v+8
    const int mhi = (lane >> 4) << 3;
#pragma unroll
    for (int m = 0; m < 4; ++m) {
#pragma unroll
        for (int v = 0; v < 8; ++v) {
            int row = m * 16 + v + mhi;
            int col = n0 + ln;
            atomAddF(&out2[row * NDIM + col], acc[m][v]);
        }
    }
}

extern "C" void kernel_launch(void* const* d_in, const int* in_sizes, int n_in,
                              void* d_out, int out_size, void* d_ws, size_t ws_size,
                              hipStream_t stream) {
    const float* feat = (const float*)d_in[0];
    const int*   src  = (const int*)  d_in[1];
    const int*   dst  = (const int*)  d_in[2];
    const float* ws1  = (const float*)d_in[3];
    const float* wn1  = (const float*)d_in[4];
    const float* b1   = (const float*)d_in[5];
    const float* ws2  = (const float*)d_in[6];
    const float* wn2  = (const float*)d_in[7];
    const float* b2   = (const float*)d_in[8];
    const float* w3   = (const float*)d_in[9];
    const float* b3   = (const float*)d_in[10];
    float* out = (float*)d_out;

    const int N = in_sizes[0];      // 262144 nodes (features are [N,1])
    const int E = in_sizes[1];      // 2097152 edges
    const int H = in_sizes[5];      // 128 hidden

    float* ws     = (float*)d_ws;
    float* agg1   = ws;
    float* deg    = ws + 1 * (size_t)N;
    float* sneigh = ws + 2 * (size_t)N;
    float* agg2   = ws + 3 * (size_t)N;
    float* sself  = ws + 4 * (size_t)N;
    float* h2     = ws + 5 * (size_t)N;   // 64*4096 == N

    float* out1 = out;              // [64,4096] flattened
    float* out2 = out + N;          // [64,4096] flattened

    const int TB = 256;
    const int E4 = (E + 3) / 4;     // edges processed 4 per thread
    gpn_init_kernel <<<(N + TB - 1) / TB, TB, 0, stream>>>(agg1, deg, agg2, out2, b3, N);
    gpn_edge1_kernel<<<(E4 + TB - 1) / TB, TB, 0, stream>>>(src, dst, feat, agg1, deg, E);
    gpn_node1_kernel<<<(N + TB - 1) / TB, TB, 0, stream>>>(feat, agg1, deg,
                                                           ws1, wn1, b1, ws2, wn2, b2,
                                                           sself, sneigh, N, H);
    gpn_edge2_kernel<<<(E4 + TB - 1) / TB, TB, 0, stream>>>(src, dst, sneigh, agg2, E);
    gpn_node2_kernel<<<(N + TB - 1) / TB, TB, 0, stream>>>(sself, agg2, deg, out1, h2, N);
    // 64 N-strip groups x KSPLIT K-chunks = 512 blocks, 4 waves each
    gpn_gemm_kernel <<<dim3(NDIM / 64, KSPLIT), 128, 0, stream>>>(h2, w3, out2);
}